// CgTransform_19292993094300
// MI455X (gfx1250) — compile-verified
//
#include <hip/hip_runtime.h>

#define FDIM 1024
#define F2   (1024*1024)

typedef __attribute__((ext_vector_type(2))) float v2f;
typedef __attribute__((ext_vector_type(4))) float v4f;
typedef __attribute__((ext_vector_type(8))) float v8f;

// Triplet tables in reference cgc order: (l1,l2,l) and block position within per_l[l]
__constant__ int t_l1[23]   = {0,0,1,1,1,0,1,1,1,2,2,2,2,0,1,1,2,2,2,3,3,3,3};
__constant__ int t_l2[23]   = {0,1,1,1,1,2,2,2,2,2,2,2,2,3,3,3,3,3,3,3,3,3,3};
__constant__ int t_l [23]   = {0,1,0,1,2,2,1,2,3,0,1,2,3,3,2,3,1,2,3,0,1,2,3};
__constant__ int t_bpos[23] = {0,0,1,1,0,1,2,3,0,2,3,3,1,2,4,3,4,5,4,3,5,6,5};
__constant__ int c_nblk[4]  = {4,6,7,6};   // blocks concatenated per output l
__constant__ int c_lbase[4] = {0,4,22,57}; // row offset (units of F^2) of per_l[l]

struct Args {
  const float* clms;
  float*       out;
  const float* cg[23];
};

// One block handles plane p = (triplet cgc, output row m) and an 8-tile-wide
// column group. 8 waves/block, each wave owns one 16-col g-stripe and loops
// over all 64 16-row f-tiles, reusing its B fragment (= C-row contracted with
// f2 columns). D(16x16) = A(16x4:f1^T) x B(4x16:T) accumulated over K<=8.
__global__ __launch_bounds__(256) void cg_wmma_kernel(Args args) {
  __shared__ float sf1[8 * FDIM];   // f1 rows, zero-padded to K=8 (32 KB)

  // plane -> (triplet cgc, m); force scalar so downstream branches are uniform
  int cgc = 0, m = blockIdx.x;
  while (m >= 2 * t_l[cgc] + 1) { m -= 2 * t_l[cgc] + 1; ++cgc; }
  cgc = __builtin_amdgcn_readfirstlane(cgc);
  m   = __builtin_amdgcn_readfirstlane(m);

  const int l1 = t_l1[cgc], l2 = t_l2[cgc], l = t_l[cgc];
  const int n1 = 2 * l1 + 1, n2 = 2 * l2 + 1;

  const float* f1   = args.clms + (size_t)(l1 * l1) * FDIM;
  const float* f2   = args.clms + (size_t)(l2 * l2) * FDIM;
  const float* Crow = args.cg[cgc] + (size_t)m * n1 * n2;
  float* outp = args.out +
      (size_t)(c_lbase[l] + m * c_nblk[l] + t_bpos[cgc]) * (size_t)F2;

  // Cooperative stage of f1 (n1 contiguous rows of clms) into LDS, zero-pad to 8 rows.
  {
    const v4f* src = (const v4f*)f1;
    v4f*       dst = (v4f*)sf1;
    const int nvalid = n1 * (FDIM / 4);
    for (int i = threadIdx.x; i < 8 * (FDIM / 4); i += blockDim.x) {
      v4f v = {0.f, 0.f, 0.f, 0.f};
      if (i < nvalid) v = src[i];
      dst[i] = v;
    }
  }
  __syncthreads();

  const int lane = threadIdx.x & 31;
  const int wave = threadIdx.x >> 5;
  const int lo   = lane & 15;       // N / M position within half
  const int hf   = lane >> 4;       // lane half
  const int g0   = (blockIdx.y * 8 + wave) * 16;
  const int k0   = 2 * hf;          // K index of vgpr0 element (ISA 16x4 f32 layout)

  // --- B fragments: B[k][c] = sum_j Crow[k*n2 + j] * f2[j, g0+c], K-padded with 0.
  float f2v[7];
#pragma unroll
  for (int j = 0; j < 7; ++j)
    f2v[j] = (j < n2) ? f2[(size_t)j * FDIM + g0 + lo] : 0.f;

  v2f B1 = {0.f, 0.f}, B2 = {0.f, 0.f};
#pragma unroll
  for (int vg = 0; vg < 2; ++vg) {
    const int ka = k0 + vg;       // K 0..3
    const int kb = k0 + vg + 4;   // K 4..7
    float acc_a = 0.f, acc_b = 0.f;
    if (ka < n1)
      for (int j = 0; j < n2; ++j) acc_a += Crow[ka * n2 + j] * f2v[j];
    if (kb < n1)
      for (int j = 0; j < n2; ++j) acc_b += Crow[kb * n2 + j] * f2v[j];
    B1[vg] = acc_a;
    B2[vg] = acc_b;
  }

  if (n1 > 4) {
    // K in (4,8]: two chained WMMAs per 16x16 tile
    for (int f0 = 0; f0 < FDIM / 16; ++f0) {
      const int abase = f0 * 16 + lo;
      v2f A1, A2;                             // A[M=lo, K] = f1[K, f0*16+lo]
      A1[0] = sf1[(k0    ) * FDIM + abase];
      A1[1] = sf1[(k0 + 1) * FDIM + abase];
      A2[0] = sf1[(k0 + 4) * FDIM + abase];
      A2[1] = sf1[(k0 + 5) * FDIM + abase];

      v8f acc = {0.f, 0.f, 0.f, 0.f, 0.f, 0.f, 0.f, 0.f};
      acc = __builtin_amdgcn_wmma_f32_16x16x4_f32(
          false, A1, false, B1, (short)0, acc, false, false);
      acc = __builtin_amdgcn_wmma_f32_16x16x4_f32(
          false, A2, false, B2, (short)0, acc, false, false);

      float* rp = outp + (size_t)(f0 * 16 + 8 * hf) * FDIM + (g0 + lo);
#pragma unroll
      for (int v = 0; v < 8; ++v)
        __builtin_nontemporal_store(acc[v], rp + v * FDIM);
    }
  } else {
    // K <= 4: single WMMA per tile, no A2/B2 needed
    for (int f0 = 0; f0 < FDIM / 16; ++f0) {
      const int abase = f0 * 16 + lo;
      v2f A1;
      A1[0] = sf1[(k0    ) * FDIM + abase];
      A1[1] = sf1[(k0 + 1) * FDIM + abase];

      v8f acc = {0.f, 0.f, 0.f, 0.f, 0.f, 0.f, 0.f, 0.f};
      acc = __builtin_amdgcn_wmma_f32_16x16x4_f32(
          false, A1, false, B1, (short)0, acc, false, false);

      float* rp = outp + (size_t)(f0 * 16 + 8 * hf) * FDIM + (g0 + lo);
#pragma unroll
      for (int v = 0; v < 8; ++v)
        __builtin_nontemporal_store(acc[v], rp + v * FDIM);
    }
  }
}

static const int cg_sizes[23] = {1, 9, 9, 27, 45, 25, 45, 75, 105, 25, 75, 125,
                                 175, 49, 105, 147, 105, 175, 245, 49, 147, 245, 343};

extern "C" void kernel_launch(void* const* d_in, const int* in_sizes, int n_in,
                              void* d_out, int out_size, void* d_ws, size_t ws_size,
                              hipStream_t stream) {
  Args a;
  a.clms = (const float*)d_in[0];
  a.out  = (float*)d_out;
  if (n_in >= 24) {                       // tuple flattened to separate inputs
    for (int i = 0; i < 23; ++i) a.cg[i] = (const float*)d_in[1 + i];
  } else {                                // tuple concatenated into one buffer
    const float* base = (const float*)d_in[1];
    size_t off = 0;
    for (int i = 0; i < 23; ++i) { a.cg[i] = base + off; off += cg_sizes[i]; }
  }
  dim3 grid(99, 8, 1), block(256, 1, 1);
  hipLaunchKernelGGL(cg_wmma_kernel, grid, block, 0, stream, a);
  (void)in_sizes; (void)out_size; (void)d_ws; (void)ws_size;
}